// NeuralODE_8787503087638
// MI455X (gfx1250) — compile-verified
//
#include <hip/hip_runtime.h>

typedef float v2f __attribute__((ext_vector_type(2)));
typedef float v8f __attribute__((ext_vector_type(8)));

#define LSZ   2048
#define CH    4
#define HID   64
#define STRIPW 128
#define NSTRIP 16
#define NT    9          // tiles of 16 covering strip + halo (144 cols)
#define WPAD  148        // padded row stride (floats) for hid/gbuf -> conflict-free banks
#define GROWS 20

__device__ __forceinline__ float fast_tanh(float x) {
#if __has_builtin(__builtin_amdgcn_tanhf)
    return __builtin_amdgcn_tanhf(x);           // v_tanh_f32
#else
    // tanh(x) = 1 - 2/(e^{2x}+1), e^{2x} = 2^{2x*log2(e)}
    float t = __builtin_amdgcn_exp2f(x * 2.8853900817779268f);
    return 1.0f - 2.0f * __builtin_amdgcn_rcpf(t + 1.0f);
#endif
}

__device__ __forceinline__ int clampi(int v, int lo, int hi) {
    return v < lo ? lo : (v > hi ? hi : v);
}

__global__ __launch_bounds__(256)
void node_rk3_kernel(const float* __restrict__ Z0,
                     const float* __restrict__ W1g,
                     const float* __restrict__ B1g,
                     const float* __restrict__ W2g,
                     const float* __restrict__ B2g,
                     const int*   __restrict__ Tg,
                     float* __restrict__ OUT)
{
    __shared__ float zb[3][CH * LSZ];          // 96 KB: z0 / k-stage ping-pong
    __shared__ float hid[HID * WPAD];          // 37 KB: tanh(conv1) strip
    __shared__ float gbuf[GROWS * WPAD];       // 11.6 KB: packed conv2 partials

    const int tid  = threadIdx.x;
    const int lane = tid & 31;
    const int ln   = lane & 15;    // N / M index within tile
    const int lh   = lane >> 4;    // K-half select
    const int wave = tid >> 5;
    const size_t base = (size_t)blockIdx.x * (CH * LSZ);

    // ---- weight fragments in registers (persist whole kernel) ----
    // conv1: A[h][k], k = cin*5+dk, M=64 (4 tiles), K=20 (5 k-tiles)
    v2f a1[4][5];
    int c1cin[5][2], c1yof[5][2];
    #pragma unroll
    for (int kt = 0; kt < 5; ++kt) {
        int k0 = kt * 4 + lh * 2;
        c1cin[kt][0] = (k0 / 5) * LSZ;       c1yof[kt][0] = (k0 % 5) - 2;
        c1cin[kt][1] = ((k0+1) / 5) * LSZ;   c1yof[kt][1] = ((k0+1) % 5) - 2;
        #pragma unroll
        for (int mt = 0; mt < 4; ++mt) {
            int hrow = mt * 16 + ln;
            a1[mt][kt].x = W1g[hrow * 20 + k0];
            a1[mt][kt].y = W1g[hrow * 20 + k0 + 1];
        }
    }
    // conv2 packed: A2[r][h], r = cout*5+dk (20 rows -> 2 M-tiles), K=64 (16 k-tiles)
    v2f a2[2][16];
    int hofA[16];
    #pragma unroll
    for (int mt = 0; mt < 2; ++mt) {
        int r = mt * 16 + ln;
        bool valid = (r < GROWS);
        int co = r / 5, dk = r % 5;
        #pragma unroll
        for (int kt = 0; kt < 16; ++kt) {
            int h0 = kt * 4 + lh * 2;
            if (mt == 0) hofA[kt] = h0 * WPAD;
            a2[mt][kt].x = valid ? W2g[co * 320 + h0 * 5 + dk]       : 0.0f;
            a2[mt][kt].y = valid ? W2g[co * 320 + (h0 + 1) * 5 + dk] : 0.0f;
        }
    }
    float b1r[32];
    #pragma unroll
    for (int mt = 0; mt < 4; ++mt)
        #pragma unroll
        for (int v = 0; v < 8; ++v)
            b1r[mt * 8 + v] = B1g[mt * 16 + lh * 8 + v];

    const int coutT = tid >> 6;           // combine-phase ownership
    const int xl0   = (tid & 63) * 2;
    const float b2r = B2g[coutT];
    const int gbase = coutT * 5 * WPAD;

    // ---- load z0 into LDS ----
    for (int i = tid; i < CH * LSZ; i += 256) zb[0][i] = Z0[base + i];

    const int tt     = Tg[0];
    const int nsteps = tt * 20;                    // round(t/0.05)
    const float h    = (float)tt / (float)nsteps;

    int iz = 0, ia = 1, ib = 2;                    // buffer indices
    const v8f vzero = {0.f,0.f,0.f,0.f,0.f,0.f,0.f,0.f};

    __syncthreads();

    #pragma unroll 1
    for (int step = 0; step < nsteps; ++step) {
        #pragma unroll 1
        for (int stage = 0; stage < 3; ++stage) {
            int si, so; float cA, cB, cC;
            if (stage == 0)      { si = iz; so = ia; cA = 0.0f;      cB = 1.0f;      cC = h; }
            else if (stage == 1) { si = ia; so = ib; cA = 0.75f;     cB = 0.25f;     cC = 0.25f * h; }
            else                 { si = ib; so = ia; cA = 1.f/3.f;   cB = 2.f/3.f;   cC = (2.f/3.f) * h; }
            const float* zin = zb[si];
            float*       zout = zb[so];
            const float* zst = zb[iz];

            #pragma unroll 1
            for (int strip = 0; strip < NSTRIP; ++strip) {
                const int S0 = strip * STRIPW;

                // ---- conv1 + bias + tanh -> hid strip (cols cover [S0-2, S0+142)) ----
                #pragma unroll 1
                for (int t = wave; t < NT; t += 8) {
                    const int Yb = S0 - 2 + 16 * t + ln;
                    v2f bf[5];
                    #pragma unroll
                    for (int kt = 0; kt < 5; ++kt) {
                        // branchless zero-pad: clamped address + post-load select
                        int y0  = Yb + c1yof[kt][0];
                        int y0c = clampi(y0, 0, LSZ - 1);
                        float v0 = zin[c1cin[kt][0] + y0c];
                        bf[kt].x = (y0 == y0c) ? v0 : 0.0f;
                        int y1  = Yb + c1yof[kt][1];
                        int y1c = clampi(y1, 0, LSZ - 1);
                        float v1 = zin[c1cin[kt][1] + y1c];
                        bf[kt].y = (y1 == y1c) ? v1 : 0.0f;
                    }
                    v8f acc0 = vzero, acc1 = vzero, acc2 = vzero, acc3 = vzero;
                    #pragma unroll
                    for (int kt = 0; kt < 5; ++kt) {
                        acc0 = __builtin_amdgcn_wmma_f32_16x16x4_f32(false, a1[0][kt], false, bf[kt], (short)0, acc0, false, false);
                        acc1 = __builtin_amdgcn_wmma_f32_16x16x4_f32(false, a1[1][kt], false, bf[kt], (short)0, acc1, false, false);
                        acc2 = __builtin_amdgcn_wmma_f32_16x16x4_f32(false, a1[2][kt], false, bf[kt], (short)0, acc2, false, false);
                        acc3 = __builtin_amdgcn_wmma_f32_16x16x4_f32(false, a1[3][kt], false, bf[kt], (short)0, acc3, false, false);
                    }
                    const int pcol = 16 * t + ln;
                    #pragma unroll
                    for (int v = 0; v < 8; ++v) {
                        int rr = lh * 8 + v;
                        hid[(rr +  0) * WPAD + pcol] = fast_tanh(acc0[v] + b1r[ 0 + v]);
                        hid[(rr + 16) * WPAD + pcol] = fast_tanh(acc1[v] + b1r[ 8 + v]);
                        hid[(rr + 32) * WPAD + pcol] = fast_tanh(acc2[v] + b1r[16 + v]);
                        hid[(rr + 48) * WPAD + pcol] = fast_tanh(acc3[v] + b1r[24 + v]);
                    }
                }
                __syncthreads();

                // ---- conv2 partials: G[r=(cout,dk)][p] = sum_h W2p[r][h]*hid[h][p] ----
                #pragma unroll 1
                for (int nt = wave; nt < NT; nt += 8) {
                    const int p = nt * 16 + ln;
                    const int y = S0 - 2 + p;
                    const bool inr = (y >= 0 && y < LSZ);   // conv2 zero-padding
                    v8f g0 = vzero, g1 = vzero;
                    #pragma unroll
                    for (int kt = 0; kt < 16; ++kt) {
                        // hid[..]+p is always in-bounds of the LDS strip; zero-pad by select
                        float vx = hid[hofA[kt] + p];
                        float vy = hid[hofA[kt] + WPAD + p];
                        v2f bh;
                        bh.x = inr ? vx : 0.0f;
                        bh.y = inr ? vy : 0.0f;
                        g0 = __builtin_amdgcn_wmma_f32_16x16x4_f32(false, a2[0][kt], false, bh, (short)0, g0, false, false);
                        g1 = __builtin_amdgcn_wmma_f32_16x16x4_f32(false, a2[1][kt], false, bh, (short)0, g1, false, false);
                    }
                    #pragma unroll
                    for (int v = 0; v < 8; ++v) {
                        int r0 = lh * 8 + v;
                        gbuf[r0 * WPAD + p] = g0[v];
                        int r1 = 16 + lh * 8 + v;
                        if (r1 < GROWS) gbuf[r1 * WPAD + p] = g1[v];
                    }
                }
                __syncthreads();

                // ---- combine: zout = cA*z0 + cB*zin + cC*(b2 + sum_dk G[cout*5+dk][xl+dk]) ----
                #pragma unroll
                for (int j = 0; j < 2; ++j) {
                    int xl = xl0 + j;
                    float f = b2r;
                    #pragma unroll
                    for (int dk = 0; dk < 5; ++dk)
                        f += gbuf[gbase + dk * (WPAD + 1) + xl];
                    int idx = coutT * LSZ + S0 + xl;
                    zout[idx] = cA * zst[idx] + cB * zin[idx] + cC * f;
                }
                // no barrier needed here: next strip's post-conv1 barrier orders gbuf reuse
            }

            // ---- periodic ghost refresh (_bdry) on zout ----
            __syncthreads();
            if (tid < 80) {
                int c = tid / 20, i = tid % 20;
                if (i < 10) zout[c * LSZ + i]        = zout[c * LSZ + 2028 + i];
                else        zout[c * LSZ + 2028 + i] = zout[c * LSZ + i];
            }
            __syncthreads();
        }
        // z_new is in buffer ia after stage 3 -> rotate
        int tmp = iz; iz = ia; ia = tmp;
    }

    for (int i = tid; i < CH * LSZ; i += 256) OUT[base + i] = zb[iz][i];
}

extern "C" void kernel_launch(void* const* d_in, const int* in_sizes, int n_in,
                              void* d_out, int out_size, void* d_ws, size_t ws_size,
                              hipStream_t stream) {
    const float* z0 = (const float*)d_in[0];
    const float* W1 = (const float*)d_in[1];
    const float* b1 = (const float*)d_in[2];
    const float* W2 = (const float*)d_in[3];
    const float* b2 = (const float*)d_in[4];
    const int*   t1 = (const int*)d_in[5];
    float* out = (float*)d_out;
    int m = in_sizes[0] / (CH * LSZ);   // 32 independent samples
    node_rk3_kernel<<<m, 256, 0, stream>>>(z0, W1, b1, W2, b2, t1, out);
}